// PrototypeGCN_3049426780611
// MI455X (gfx1250) — compile-verified
//
#include <hip/hip_runtime.h>
#include <math.h>

// GCN 2-layer forward for MI455X (gfx1250, wave32).
// D = 128 fixed by the reference.
#define DIM 128

typedef __attribute__((ext_vector_type(2))) float v2f;
typedef __attribute__((ext_vector_type(8))) float v8f;

// ---------------------------------------------------------------------------
// small elementwise helpers
// ---------------------------------------------------------------------------
__global__ void __launch_bounds__(256) k_fill(float* __restrict__ p, float v, int n) {
    int i = blockIdx.x * blockDim.x + threadIdx.x;
    if (i < n) p[i] = v;
}

__global__ void __launch_bounds__(256) k_deg(float* __restrict__ deg,
                                             const int* __restrict__ col, int E) {
    int i = blockIdx.x * blockDim.x + threadIdx.x;
    if (i < E) {
        (void)__hip_atomic_fetch_add(&deg[col[i]], 1.0f,
                                     __ATOMIC_RELAXED, __HIP_MEMORY_SCOPE_AGENT);
    }
}

__global__ void __launch_bounds__(256) k_rsqrt(float* __restrict__ d, int n) {
    int i = blockIdx.x * blockDim.x + threadIdx.x;
    if (i < n) d[i] = rsqrtf(d[i]);   // deg >= 1 always (self loop), no zero guard needed
}

// agg[i][c] = bias[c]  (pre-seeds the scatter accumulator with the layer bias)
__global__ void __launch_bounds__(256) k_bias_init(float* __restrict__ agg,
                                                   const float* __restrict__ b, int nd) {
    int i = blockIdx.x * blockDim.x + threadIdx.x;
    if (i < nd) agg[i] = b[i & (DIM - 1)];
}

__global__ void __launch_bounds__(256) k_relu(float* __restrict__ p, int n) {
    int i = blockIdx.x * blockDim.x + threadIdx.x;
    if (i < n) p[i] = fmaxf(p[i], 0.0f);
}

// ---------------------------------------------------------------------------
// Y = X @ W^T   (X: [N,128], W: [128,128] row-major, Y: [N,128])
// fp32 WMMA 16x16x4. One wave -> one 16x16 tile of Y; 8 waves/block cover all
// 128 output columns of a 16-row strip. A-tile staged in LDS, shared by waves.
// RELU fuses max(x,0) into the A load (layer-2 reads layer-1 activations).
// ---------------------------------------------------------------------------
template <bool RELU>
__global__ void __launch_bounds__(256) k_gemm_wmma(const float* __restrict__ X,
                                                   const float* __restrict__ W,
                                                   float* __restrict__ Y, int N) {
    __shared__ float xs[16 * DIM];                 // 8 KB A-tile
    const int t     = threadIdx.x;
    const int mBase = blockIdx.x * 16;

    // --- stage 16x128 tile of X into LDS (each thread: 8 contiguous floats) ---
    {
        int row  = t >> 4;                         // 0..15
        int colc = (t & 15) * 8;                   // 0,8,...,120
        int mrow = mBase + row; if (mrow >= N) mrow = N - 1;   // clamp, no divergence
        const float* src = X + (size_t)mrow * DIM + colc;
        float4 a = *(const float4*)(src);
        float4 b = *(const float4*)(src + 4);
        if (RELU) {
            a.x = fmaxf(a.x, 0.f); a.y = fmaxf(a.y, 0.f);
            a.z = fmaxf(a.z, 0.f); a.w = fmaxf(a.w, 0.f);
            b.x = fmaxf(b.x, 0.f); b.y = fmaxf(b.y, 0.f);
            b.z = fmaxf(b.z, 0.f); b.w = fmaxf(b.w, 0.f);
        }
        *(float4*)&xs[row * DIM + colc]     = a;
        *(float4*)&xs[row * DIM + colc + 4] = b;
    }
    __syncthreads();

    const int lane  = t & 31;
    const int l     = lane & 15;                   // row-in-tile / col-in-tile
    const int half  = lane >> 4;                   // 0: K pair {k,k+1}, 1: {k+2,k+3}
    const int nBase = (t >> 5) * 16;               // wave -> 16 output columns

    // B[k][n] = W[n][k]  -> per-lane fragment is a contiguous float2 of row n of W
    const float* wrow = W + (size_t)(nBase + l) * DIM;
    const float* arow = &xs[l * DIM];

    v8f c = {};                                    // 16x16 f32 accumulator (8 VGPRs)
#pragma unroll
    for (int k0 = 0; k0 < DIM; k0 += 4) {
        const int k = k0 + 2 * half;
        v2f a; a.x = arow[k]; a.y = arow[k + 1];   // A 16x4 fp32 fragment (ISA layout)
        v2f b; b.x = wrow[k]; b.y = wrow[k + 1];   // B 4x16 fp32 fragment
        c = __builtin_amdgcn_wmma_f32_16x16x4_f32(
                /*neg_a=*/false, a, /*neg_b=*/false, b,
                /*c_mod=*/(short)0, c, /*reuse_a=*/false, /*reuse_b=*/false);
    }

    // C/D layout: lanes 0-15: M = v, lanes 16-31: M = v + 8; N = l
    const int mRow = mBase + half * 8;
    float* yp = Y + (size_t)mRow * DIM + nBase + l;
#pragma unroll
    for (int v = 0; v < 8; ++v) {
        if (mRow + v < N) yp[(size_t)v * DIM] = c[v];
    }
}

// ---------------------------------------------------------------------------
// out[col] += dis[row]*dis[col] * Y[row]   (one wave per edge; lane = float4
// slice of the 128-wide row; native f32 atomics into L2)
// ---------------------------------------------------------------------------
__global__ void __launch_bounds__(256) k_aggregate(const float* __restrict__ Y,
                                                   const float* __restrict__ dis,
                                                   const int* __restrict__ row,
                                                   const int* __restrict__ col,
                                                   int Etot, int E,
                                                   float* __restrict__ out) {
    const int lane   = threadIdx.x & 31;
    const int wid    = (blockIdx.x * blockDim.x + threadIdx.x) >> 5;
    const int nwaves = (gridDim.x * blockDim.x) >> 5;

    for (int e = wid; e < Etot; e += nwaves) {
        int r, c;
        if (e < E) { r = row[e]; c = col[e]; }
        else       { r = e - E;  c = e - E;  }     // self loop
        const float norm = dis[r] * dis[c];

        const float4 v = *(const float4*)(Y + (size_t)r * DIM + lane * 4);
        float* dst = out + (size_t)c * DIM + lane * 4;
        (void)__hip_atomic_fetch_add(dst + 0, norm * v.x, __ATOMIC_RELAXED, __HIP_MEMORY_SCOPE_AGENT);
        (void)__hip_atomic_fetch_add(dst + 1, norm * v.y, __ATOMIC_RELAXED, __HIP_MEMORY_SCOPE_AGENT);
        (void)__hip_atomic_fetch_add(dst + 2, norm * v.z, __ATOMIC_RELAXED, __HIP_MEMORY_SCOPE_AGENT);
        (void)__hip_atomic_fetch_add(dst + 3, norm * v.w, __ATOMIC_RELAXED, __HIP_MEMORY_SCOPE_AGENT);
    }
}

// ---------------------------------------------------------------------------
extern "C" void kernel_launch(void* const* d_in, const int* in_sizes, int n_in,
                              void* d_out, int out_size, void* d_ws, size_t ws_size,
                              hipStream_t stream) {
    const float* x    = (const float*)d_in[0];
    const float* W1   = (const float*)d_in[1];
    const float* b1   = (const float*)d_in[2];
    const float* W2   = (const float*)d_in[3];
    const float* b2   = (const float*)d_in[4];
    const int*   edge = (const int*)d_in[5];

    const int N    = in_sizes[0] / DIM;            // 100000
    const int E    = in_sizes[5] / 2;              // 640000
    const int Etot = E + N;                        // edges + self loops
    const int ND   = N * DIM;

    const int* row = edge;
    const int* col = edge + E;
    float* out = (float*)d_out;

    // workspace layout: dis[Npad] | ybuf[N*D] | agg1[N*D]
    float* ws   = (float*)d_ws;
    size_t Npad = ((size_t)N + 3) & ~(size_t)3;    // keep 16B alignment downstream
    float* dis  = ws;
    float* ybuf = ws + Npad;
    float* agg1 = ybuf + (size_t)N * DIM;

    const int TB = 256;
    const int mblocks  = (N + 15) / 16;            // 6250 (exact)
    const int eblocks  = (Etot + 7) / 8;           // 1 edge per wave, 8 waves/block

    // ---- symmetric normalization: dis = rsqrt(in_degree + 1) ----
    k_fill <<<(N + TB - 1) / TB, TB, 0, stream>>>(dis, 1.0f, N);
    k_deg  <<<(E + TB - 1) / TB, TB, 0, stream>>>(dis, col, E);
    k_rsqrt<<<(N + TB - 1) / TB, TB, 0, stream>>>(dis, N);

    // ---- layer 1: y = x @ W1^T ; agg1 = b1 + scatter(norm * y[row]) ----
    k_gemm_wmma<false><<<mblocks, TB, 0, stream>>>(x, W1, ybuf, N);
    k_bias_init<<<(ND + TB - 1) / TB, TB, 0, stream>>>(agg1, b1, ND);
    k_aggregate<<<eblocks, TB, 0, stream>>>(ybuf, dis, row, col, Etot, E, agg1);

    // ---- layer 2: y = relu(agg1) @ W2^T ; out = b2 + scatter(...) ; relu ----
    k_gemm_wmma<true><<<mblocks, TB, 0, stream>>>(agg1, W2, ybuf, N);
    k_bias_init<<<(ND + TB - 1) / TB, TB, 0, stream>>>(out, b2, ND);
    k_aggregate<<<eblocks, TB, 0, stream>>>(ybuf, dis, row, col, Etot, E, out);
    k_relu<<<(ND + TB - 1) / TB, TB, 0, stream>>>(out, ND);
}